// brain_encoder_39333310496946
// MI455X (gfx1250) — compile-verified
//
#include <hip/hip_runtime.h>

// Problem constants from the reference.
constexpr int B_ = 32;      // batch
constexpr int Q_ = 180;     // parcels / queries
constexpr int H_ = 768;     // hidden
constexpr int V_ = 19004;   // voxels

typedef __attribute__((ext_vector_type(2))) float v2f;
typedef __attribute__((ext_vector_type(8))) float v8f;

// ---------------- Pass 0-3: counting sort of voxels by parcel id ----------------

__global__ void be_zero_counts(int* __restrict__ counts) {
    int i = blockIdx.x * blockDim.x + threadIdx.x;
    if (i < Q_) counts[i] = 0;
}

__global__ void be_hist(const int* __restrict__ pid, int* __restrict__ counts) {
    int v = blockIdx.x * blockDim.x + threadIdx.x;
    if (v < V_) atomicAdd(&counts[pid[v]], 1);
}

__global__ void be_scan(const int* __restrict__ counts,
                        int* __restrict__ offsets, int* __restrict__ cursor) {
    if (threadIdx.x == 0 && blockIdx.x == 0) {
        int acc = 0;
        for (int q = 0; q < Q_; ++q) {
            offsets[q] = acc;
            cursor[q]  = acc;
            acc += counts[q];
        }
        offsets[Q_] = acc;
    }
}

__global__ void be_scatter(const int* __restrict__ pid, int* __restrict__ cursor,
                           int* __restrict__ sorted) {
    int v = blockIdx.x * blockDim.x + threadIdx.x;
    if (v < V_) {
        int q = pid[v];
        int pos = atomicAdd(&cursor[q], 1);
        sorted[pos] = v;
    }
}

// ---------------- Pass 4: per-parcel gathered GEMM via V_WMMA_F32_16X16X4_F32 ----------------
//
// Block = one parcel q, 256 threads = 8 waves. tokens[:,q,:] (32x768 f32, 96 KB)
// is staged in LDS once. Each wave owns 16-voxel tiles (stride 8): two 16x16 f32
// accumulators (batches 0-15 and 16-31) share one B tile; K walks H in steps of 4.
//
// WMMA f32 16x16x4 operand layout (ISA 7.12.2):
//   A (16x4): lane m (0-15) VGPR0/1 = K0,K1 ; lane 16+m = K2,K3   -> float2 @ h0+2*(lane>>4)
//   B (4x16): lane n (0-15) VGPR0/1 = K0,K1 ; lane 16+n = K2,K3   -> same addressing on W row
//   C/D     : VGPR r = (M=r, lanes 0-15) and (M=8+r, lanes 16-31), N = lane&15
__global__ __launch_bounds__(256) void be_parcel_gemm(
    const float* __restrict__ tokens,   // [B, Q, H]
    const float* __restrict__ W,        // [V, H]  (streamed once, non-temporal)
    const float* __restrict__ bias,     // [V]
    const int*   __restrict__ sorted,   // [V] voxel ids grouped by parcel
    const int*   __restrict__ offsets,  // [Q+1]
    float*       __restrict__ out) {    // [B, V]
    extern __shared__ float toks[];     // B_*H_ = 24576 floats = 96 KB

    const int q = blockIdx.x;

    // Stage tokens[:, q, :] into LDS (coalesced 128-bit loads).
    for (int i4 = threadIdx.x; i4 < (B_ * H_) / 4; i4 += 256) {
        const int b  = i4 / (H_ / 4);
        const int h4 = (i4 % (H_ / 4)) * 4;
        const float4 s = *reinterpret_cast<const float4*>(
            tokens + ((size_t)b * Q_ + q) * H_ + h4);
        *reinterpret_cast<float4*>(&toks[b * H_ + h4]) = s;
    }
    __syncthreads();

    const int start    = offsets[q];
    const int end      = offsets[q + 1];
    const int numTiles = (end - start + 15) >> 4;

    const int wave = threadIdx.x >> 5;
    const int lane = threadIdx.x & 31;
    const int col  = lane & 15;        // N index (voxel within tile) / M index for A
    const int khi  = lane >> 4;        // 0 -> K0,K1 ; 1 -> K2,K3

    const float* toksA0 = &toks[col * H_ + 2 * khi];         // batches 0-15
    const float* toksA1 = &toks[(16 + col) * H_ + 2 * khi];  // batches 16-31

    for (int t = wave; t < numTiles; t += 8) {               // wave-uniform loop
        const int  vi    = start + t * 16 + col;
        const bool valid = vi < end;
        const int  vox   = sorted[valid ? vi : start];       // clamp, mask at store
        const float* wrow = W + (size_t)vox * H_ + 2 * khi;

        v8f acc0 = {};   // batches 0-15
        v8f acc1 = {};   // batches 16-31

#pragma unroll 4
        for (int h0 = 0; h0 < H_; h0 += 4) {
            const v2f a0 = *reinterpret_cast<const v2f*>(toksA0 + h0);
            const v2f a1 = *reinterpret_cast<const v2f*>(toksA1 + h0);
            // W is touched exactly once -> non-temporal streaming load.
            const v2f bb = __builtin_nontemporal_load(
                reinterpret_cast<const v2f*>(wrow + h0));
            acc0 = __builtin_amdgcn_wmma_f32_16x16x4_f32(
                false, a0, false, bb, (short)0, acc0, false, false);
            acc1 = __builtin_amdgcn_wmma_f32_16x16x4_f32(
                false, a1, false, bb, (short)0, acc1, false, false);
        }

        if (valid) {
            const float bv    = bias[vox];
            const int   mbase = khi * 8;   // lanes 16-31 hold rows M=8..15
#pragma unroll
            for (int r = 0; r < 8; ++r) {
                out[(size_t)(mbase + r) * V_ + vox]      = acc0[r] + bv;
                out[(size_t)(16 + mbase + r) * V_ + vox] = acc1[r] + bv;
            }
        }
    }
}

// ---------------- Host-side launch ----------------

extern "C" void kernel_launch(void* const* d_in, const int* in_sizes, int n_in,
                              void* d_out, int out_size, void* d_ws, size_t ws_size,
                              hipStream_t stream) {
    const float* tokens = (const float*)d_in[0];   // [B,Q,H] f32
    const float* W      = (const float*)d_in[1];   // [V,H]  f32
    const float* bias   = (const float*)d_in[2];   // [V]    f32
    const int*   pid    = (const int*)  d_in[3];   // [V]    i32
    float*       out    = (float*)d_out;           // [B,V]  f32

    // Workspace layout (ints): counts[Q] | offsets[Q+1] | cursor[Q] | sorted[V]
    int* counts  = (int*)d_ws;
    int* offsets = counts + Q_;
    int* cursor  = offsets + Q_ + 1;
    int* sorted  = cursor + Q_;

    be_zero_counts<<<1, 256, 0, stream>>>(counts);
    be_hist<<<(V_ + 255) / 256, 256, 0, stream>>>(pid, counts);
    be_scan<<<1, 32, 0, stream>>>(counts, offsets, cursor);
    be_scatter<<<(V_ + 255) / 256, 256, 0, stream>>>(pid, cursor, sorted);

    const size_t ldsBytes = (size_t)B_ * H_ * sizeof(float);   // 96 KB of 320 KB/WGP
    be_parcel_gemm<<<Q_, 256, ldsBytes, stream>>>(tokens, W, bias, sorted, offsets, out);
}